// LSTMModel_86500641342196
// MI455X (gfx1250) — compile-verified
//
#include <hip/hip_runtime.h>

typedef __attribute__((ext_vector_type(2))) float v2f;
typedef __attribute__((ext_vector_type(8))) float v8f;

#define HH 50      // hidden size
#define HP 64      // hidden padded to 4 WMMA tiles
#define GP 256     // 4 gates * 64 padded columns
#define TT 512     // sequence length
#define BB 1024    // batch
#define BT 32      // batch rows per workgroup

// Branch-free activations: v_exp_f32 + v_rcp_f32, correct saturation at +/-inf.
__device__ __forceinline__ float sigmoidf_(float x) {
    return __builtin_amdgcn_rcpf(1.0f + __expf(-x));
}
__device__ __forceinline__ float tanhf_(float x) {
    return 1.0f - 2.0f * __builtin_amdgcn_rcpf(__expf(2.0f * x) + 1.0f);
}

// One fused LSTM layer: per time step computes  g = [x_t ; h] * [Wih ; Whh]^T + (bih+bhh)
// from LDS using V_WMMA_F32_16X16X4_F32, then gate nonlinearities.
// - h lives in LDS (feeds the next step's WMMA A-matrix across waves)
// - c lives entirely in registers: each (row, col) cell is owned by a fixed lane
// - x_{t+1} is double-buffered through registers so HBM latency hides under WMMAs
// - padded columns (j >= 50) have zero weights/bias -> state stays exactly 0,
//   so the update runs unconditionally (no EXEC divergence).
template<int D, int IN_P>
__global__ __launch_bounds__(256) void lstm_layer_kernel(
    const float* __restrict__ X,     // [B, T, D]
    const float* __restrict__ Wih,   // [200, D]
    const float* __restrict__ Whh,   // [200, 50]
    const float* __restrict__ bih,   // [200]
    const float* __restrict__ bhh,   // [200]
    float* __restrict__ Y,           // [B, T, 50]  (if writeY)
    float* __restrict__ Hlast,       // [B, 50]     (if !writeY)
    int writeY)
{
    constexpr int K_P = IN_P + HP;           // padded K (x part + h part)
    constexpr int NX  = (BT * IN_P) / 256;   // x elements staged per thread

    __shared__ float Wc[GP][K_P];    // combined padded weights, row = padded gate col
    __shared__ float biasS[GP];
    __shared__ float Abuf[BT][K_P];  // [x_t | h] A-matrix staging

    const int tid  = threadIdx.x;
    const int wave = tid >> 5;
    const int lane = tid & 31;
    const int lr   = lane & 15;
    const int half = lane >> 4;
    const int k0   = half * 2;       // K sub-offset for A/B fragments
    const int q    = wave & 3;       // which 16-column slice of each gate
    const int mt   = wave >> 2;      // which 16-row M tile (0..1)
    const int b0   = blockIdx.x * BT;

    // ---- one-time: stage combined padded weights + bias into LDS ----
    {
        const int gp   = tid;        // 256 threads -> 256 padded gate rows
        const int gate = gp >> 6;
        const int j    = gp & 63;
        const int grow = gate * HH + j;
        const bool valid = (j < HH);
        for (int k = 0; k < K_P; ++k) {
            float v = 0.0f;
            if (valid) {
                if (k < IN_P) { if (k < D) v = Wih[grow * D + k]; }
                else { int kk = k - IN_P; if (kk < HH) v = Whh[grow * HH + kk]; }
            }
            Wc[gp][k] = v;
        }
        biasS[gp] = valid ? (bih[grow] + bhh[grow]) : 0.0f;
    }
    // zero h region of Abuf
    for (int idx = tid; idx < BT * HP; idx += 256) {
        int r = idx / HP, j = idx % HP;
        Abuf[r][IN_P + j] = 0.0f;
    }
    // prologue: stage x_0 directly into the A-buffer
    #pragma unroll
    for (int s = 0; s < NX; ++s) {
        int idx = tid + s * 256;
        int r = idx / IN_P, d = idx % IN_P;
        Abuf[r][d] = (d < D) ? X[(size_t)(b0 + r) * TT * D + d] : 0.0f;
    }
    __syncthreads();

    // loop-invariant bias fragment for this wave's gate columns
    float bg[4];
    #pragma unroll
    for (int g = 0; g < 4; ++g) bg[g] = biasS[g * 64 + q * 16 + lr];

    const int jcol = q * 16 + lr;          // this lane's hidden column (0..63)
    const bool jvalid = (jcol < HH);

    // register-resident cell state: slot i <-> local batch row mt*16+half*8+i
    float cre[8];
    #pragma unroll
    for (int i = 0; i < 8; ++i) cre[i] = 0.0f;

    for (int t = 0; t < TT; ++t) {
        // ---- issue x_{t+1} global loads into registers (retire under WMMAs) ----
        float xr[NX];
        if (t + 1 < TT) {
            #pragma unroll
            for (int s = 0; s < NX; ++s) {
                int idx = tid + s * 256;
                int r = idx / IN_P, d = idx % IN_P;
                xr[s] = (d < D) ? X[((size_t)(b0 + r) * TT + (t + 1)) * D + d] : 0.0f;
            }
        }

        // ---- init accumulators with bias (per-column broadcast) ----
        v8f acc[4];
        #pragma unroll
        for (int g = 0; g < 4; ++g) {
            #pragma unroll
            for (int i = 0; i < 8; ++i) acc[g][i] = bg[g];
        }

        // ---- GEMM: 16x16 tile per (mt, gate, q) via f32 WMMA, K in steps of 4 ----
        #pragma unroll 4
        for (int kt = 0; kt < K_P / 4; ++kt) {
            v2f a = *(const v2f*)&Abuf[mt * 16 + lr][kt * 4 + k0];
            #pragma unroll
            for (int g = 0; g < 4; ++g) {
                v2f b = *(const v2f*)&Wc[(g * 4 + q) * 16 + lr][kt * 4 + k0];
                acc[g] = __builtin_amdgcn_wmma_f32_16x16x4_f32(
                    false, a, false, b, (short)0, acc[g], false, false);
            }
        }
        __syncthreads();   // all WMMA reads of Abuf done before h/x are overwritten

        // ---- gate nonlinearities + state update (uniform over padded cols) ----
        #pragma unroll
        for (int i = 0; i < 8; ++i) {
            int row = mt * 16 + half * 8 + i;   // local batch row (C/D layout)
            float iv = sigmoidf_(acc[0][i]);
            float fv = sigmoidf_(acc[1][i]);
            float gv = tanhf_(acc[2][i]);
            float ov = sigmoidf_(acc[3][i]);
            float c  = fv * cre[i] + iv * gv;   // padded cols stay exactly 0
            cre[i] = c;
            float hv = ov * tanhf_(c);
            Abuf[row][IN_P + jcol] = hv;        // h for next step
            if (jvalid) {                       // masked global store only
                if (writeY) {
                    Y[((size_t)(b0 + row) * TT + t) * HH + jcol] = hv;
                } else if (t == TT - 1) {
                    Hlast[(size_t)(b0 + row) * HH + jcol] = hv;
                }
            }
        }
        // commit prefetched x_{t+1} to the A-buffer x region
        if (t + 1 < TT) {
            #pragma unroll
            for (int s = 0; s < NX; ++s) {
                int idx = tid + s * 256;
                int r = idx / IN_P, d = idx % IN_P;
                Abuf[r][d] = xr[s];
            }
        }
        __syncthreads();   // h + x staged for the next step's WMMA reads
    }
}

__global__ __launch_bounds__(256) void fc_kernel(
    const float* __restrict__ hlast,  // [B, 50]
    const float* __restrict__ fcW,    // [1, 50]
    const float* __restrict__ fcb,    // [1]
    float* __restrict__ out)          // [B, 1]
{
    int b = blockIdx.x * blockDim.x + threadIdx.x;
    if (b < BB) {
        float s = fcb[0];
        #pragma unroll
        for (int j = 0; j < HH; ++j) s += hlast[b * HH + j] * fcW[j];
        out[b] = s;
    }
}

extern "C" void kernel_launch(void* const* d_in, const int* in_sizes, int n_in,
                              void* d_out, int out_size, void* d_ws, size_t ws_size,
                              hipStream_t stream) {
    const float* x    = (const float*)d_in[0];
    const float* Wih1 = (const float*)d_in[1];
    const float* Whh1 = (const float*)d_in[2];
    const float* bih1 = (const float*)d_in[3];
    const float* bhh1 = (const float*)d_in[4];
    const float* Wih2 = (const float*)d_in[5];
    const float* Whh2 = (const float*)d_in[6];
    const float* bih2 = (const float*)d_in[7];
    const float* bhh2 = (const float*)d_in[8];
    const float* Wih3 = (const float*)d_in[9];
    const float* Whh3 = (const float*)d_in[10];
    const float* bih3 = (const float*)d_in[11];
    const float* bhh3 = (const float*)d_in[12];
    const float* fcW  = (const float*)d_in[13];
    const float* fcb  = (const float*)d_in[14];

    float* ws0   = (float*)d_ws;                    // [B, T, 50] layer1 out
    float* ws1   = ws0 + (size_t)BB * TT * HH;      // [B, T, 50] layer2 out
    float* hlast = ws1 + (size_t)BB * TT * HH;      // [B, 50]

    dim3 grid(BB / BT), block(256);
    lstm_layer_kernel<5, 16><<<grid, block, 0, stream>>>(
        x,   Wih1, Whh1, bih1, bhh1, ws0, hlast, 1);
    lstm_layer_kernel<50, 64><<<grid, block, 0, stream>>>(
        ws0, Wih2, Whh2, bih2, bhh2, ws1, hlast, 1);
    lstm_layer_kernel<50, 64><<<grid, block, 0, stream>>>(
        ws1, Wih3, Whh3, bih3, bhh3, ws0, hlast, 0);
    fc_kernel<<<dim3(4), block, 0, stream>>>(hlast, fcW, fcb, (float*)d_out);
}